// WavefunctionEmbedding_19825569038952
// MI455X (gfx1250) — compile-verified
//
#include <hip/hip_runtime.h>
#include <hip/hip_bf16.h>
#include <math.h>

// Problem constants (match reference)
#define NN   1024   // residues
#define NWL  128    // wavenumbers
#define DM   256    // d_model = 2*NWL
#define TT   16     // targets per block (WMMA M)
#define FT   16     // freqs per block
#define CH   256    // source chunk length
#define NCH  (NN / CH)

#define INV_4PI 0.07957747154594767f
#define EPS_R   1e-6f
#define LN_EPS  1e-5f

typedef __attribute__((ext_vector_type(2))) float v2f;
typedef __attribute__((ext_vector_type(8))) float v8f;

// ---------------------------------------------------------------------------
// Kernel 1: wavefunction embedding.
// grid = (NN/TT, NWL/FT) = (64, 8); block = 256 threads (8 waves).
// Each block: 16 targets x 16 freqs, streaming 4 chunks of 256 sources.
// Distances via V_WMMA_F32_16X16X4_F32 Gram tiles:  r^2 = |t|^2 + |s|^2 - 2 t.s
// Per-pair epilogue uses ONE native v_rsq_f32:  r = r2*rsq,  w = rsq/(4*pi).
// ---------------------------------------------------------------------------
__launch_bounds__(256)
__global__ void wf_embed_kernel(const float* __restrict__ coords,       // [NN*3]
                                const float* __restrict__ wavenumbers,  // [NWL]
                                const unsigned char* __restrict__ mask, // [NN] bool
                                float* __restrict__ out) {              // [NN*DM]
  const int tile = blockIdx.x * TT;   // first target of this block
  const int fgrp = blockIdx.y * FT;   // first frequency of this block
  const int t    = threadIdx.x;
  const int lane = t & 31;
  const int wid  = t >> 5;            // wave id 0..7
  const int m_acc = t >> 4;           // target row this thread accumulates (0..15)
  const int fi    = t & 15;           // freq within group (0..15)

  __shared__ float  s_tc[TT * 4];     // target coords, K-padded (x,y,z,0)
  __shared__ float  s_tn2[TT];        // |t|^2
  __shared__ float  s_sc[CH * 4];     // source coords, K-padded
  __shared__ float  s_sn2[CH];        // |s|^2
  __shared__ float  s_sval[CH];       // 1.0 if source valid (not padded)
  __shared__ float2 s_rw[TT * CH];    // interleaved (r, w) for this chunk (32KB)

  if (t < TT) {
    const int i = tile + t;
    const float x = coords[3 * i + 0];
    const float y = coords[3 * i + 1];
    const float z = coords[3 * i + 2];
    s_tc[t * 4 + 0] = x;
    s_tc[t * 4 + 1] = y;
    s_tc[t * 4 + 2] = z;
    s_tc[t * 4 + 3] = 0.0f;
    s_tn2[t] = x * x + y * y + z * z;
  }

  const float kf = wavenumbers[fgrp + fi];
  float acc_re = 0.0f, acc_im = 0.0f;

  __syncthreads();

  // Wave-resident A matrix (16x4 f32): lane l holds M = l%16, K = (l/16)*2 + v
  v2f a;
  a.x = s_tc[(lane & 15) * 4 + ((lane >> 4) * 2) + 0];
  a.y = s_tc[(lane & 15) * 4 + ((lane >> 4) * 2) + 1];

  // Preload the 8 target norms this lane's D-tile rows need (per C/D layout:
  // lane half selects rows 0..7 or 8..15).
  const int hbase = (lane >> 4) << 3;
  float tn2r[8];
  #pragma unroll
  for (int v = 0; v < 8; ++v) tn2r[v] = s_tn2[hbase + v];

  for (int c = 0; c < NCH; ++c) {
    const int base = c * CH;

    // Stage this chunk's source coords / norms / validity into LDS.
    {
      const int j = base + t;
      const float x = coords[3 * j + 0];
      const float y = coords[3 * j + 1];
      const float z = coords[3 * j + 2];
      s_sc[t * 4 + 0] = x;
      s_sc[t * 4 + 1] = y;
      s_sc[t * 4 + 2] = z;
      s_sc[t * 4 + 3] = 0.0f;
      s_sn2[t]  = x * x + y * y + z * z;
      s_sval[t] = mask[j] ? 0.0f : 1.0f;
      if (base + CH + t < NN)  // prefetch next chunk (global_prefetch_b8)
        __builtin_prefetch(&coords[3 * (base + CH + t)], 0, 1);
    }
    __syncthreads();

    // 16 source sub-tiles of 16; each wave computes 2 WMMA Gram tiles.
    #pragma unroll
    for (int s = 0; s < 2; ++s) {
      const int nn = (wid * 2 + s) * 16 + (lane & 15);  // chunk-local source col
      v2f b;
      b.x = s_sc[nn * 4 + ((lane >> 4) * 2) + 0];
      b.y = s_sc[nn * 4 + ((lane >> 4) * 2) + 1];
      v8f cz = {};
      // D[m][n] = sum_k A[m][k] * B[k][n]  (16x16x4, f32)
      v8f d = __builtin_amdgcn_wmma_f32_16x16x4_f32(
          false, a, false, b, (short)0, cz, false, false);

      const float sn2  = s_sn2[nn];
      const float sval = s_sval[nn];
      const int   j    = base + nn;
      #pragma unroll
      for (int v = 0; v < 8; ++v) {
        const int m = hbase + v;                        // D row per C/D layout
        float r2 = tn2r[v] + sn2 - 2.0f * d[v];
        r2 = fmaxf(r2, 1e-12f);                         // keep rsq finite
        const float rsq = __builtin_amdgcn_rsqf(r2);    // single v_rsq_f32
        const float r   = r2 * rsq;                     // = sqrt(r2)
        // Exclude self-pair explicitly (Gram trick can't produce exact r=0),
        // plus the reference's r>EPS cutoff and source padding mask.
        const bool  valid = (j != tile + m) && (sval != 0.0f) && (r > EPS_R);
        const float w = valid ? rsq * INV_4PI : 0.0f;   // 1/(4*pi*r), branch-free
        s_rw[m * CH + nn] = make_float2(r, w);
      }
    }
    __syncthreads();

    // Accumulate: one sin + one cos per (pair, freq) — the transcendental floor.
    const float2* __restrict__ rw = &s_rw[m_acc * CH];
    #pragma unroll 8
    for (int cn = 0; cn < CH; ++cn) {
      const float2 p  = rw[cn];
      const float  ph = kf * p.x;
      acc_re = fmaf(__cosf(ph), p.y, acc_re);
      acc_im = fmaf(__sinf(ph), p.y, acc_im);
    }
    __syncthreads();
  }

  const int i = tile + m_acc;
  out[i * DM + fgrp + fi]       = acc_re;   // real block  [0..NWL)
  out[i * DM + NWL + fgrp + fi] = acc_im;   // imag block  [NWL..2*NWL)
}

// ---------------------------------------------------------------------------
// Kernel 2: row LayerNorm (in place on d_out). One block per row.
// Matches reference: wf zeroed for padded targets BEFORE the norm; eps=1e-5.
// ---------------------------------------------------------------------------
__launch_bounds__(256)
__global__ void ln_kernel(const unsigned char* __restrict__ mask,
                          const float* __restrict__ gamma,
                          const float* __restrict__ beta,
                          float* __restrict__ out) {
  const int i = blockIdx.x;
  const int t = threadIdx.x;
  __shared__ float red[DM];

  const float x = mask[i] ? 0.0f : out[i * DM + t];

  red[t] = x;
  __syncthreads();
  #pragma unroll
  for (int s = DM / 2; s > 0; s >>= 1) {
    if (t < s) red[t] += red[t + s];
    __syncthreads();
  }
  const float mu = red[0] * (1.0f / DM);
  __syncthreads();

  const float dx = x - mu;
  red[t] = dx * dx;
  __syncthreads();
  #pragma unroll
  for (int s = DM / 2; s > 0; s >>= 1) {
    if (t < s) red[t] += red[t + s];
    __syncthreads();
  }
  const float var = red[0] * (1.0f / DM);

  const float inv_sigma = __builtin_amdgcn_rsqf(var + LN_EPS);  // v_rsq_f32
  out[i * DM + t] = dx * inv_sigma * gamma[t] + beta[t];
}

// ---------------------------------------------------------------------------
extern "C" void kernel_launch(void* const* d_in, const int* in_sizes, int n_in,
                              void* d_out, int out_size, void* d_ws, size_t ws_size,
                              hipStream_t stream) {
  (void)in_sizes; (void)n_in; (void)out_size; (void)d_ws; (void)ws_size;

  const float*         coords = (const float*)d_in[0];         // [1,1024,3]
  const float*         wn     = (const float*)d_in[1];         // [128]
  const float*         gamma  = (const float*)d_in[2];         // [256]
  const float*         beta   = (const float*)d_in[3];         // [256]
  const unsigned char* mask   = (const unsigned char*)d_in[4]; // [1,1024] bool
  float*               out    = (float*)d_out;                 // [1,1024,256]

  dim3 grid1(NN / TT, NWL / FT);   // (64, 8)
  wf_embed_kernel<<<grid1, 256, 0, stream>>>(coords, wn, mask, out);
  ln_kernel<<<NN, 256, 0, stream>>>(mask, gamma, beta, out);
}